// LSTM_84215718740627
// MI455X (gfx1250) — compile-verified
//
#include <hip/hip_runtime.h>
#include <math.h>

// ---------------- problem constants ----------------
#define BATCH 64
#define SEQ   2048
#define ISZ   256
#define HSZ   512
#define G4    (4 * HSZ)          // 2048 gate columns, order i,f,g,o

#define NWG        32            // one workgroup per 16-wide hidden column block
#define WG_THREADS 128           // 4 wave32s; wave w owns batch rows 16w..16w+15

typedef __attribute__((ext_vector_type(2))) float v2f;
typedef __attribute__((ext_vector_type(8))) float v8f;

// LDS: [gate][k][n16] for W_hh (4*512*16) then W_ih (4*256*16) = 49152 floats = 192 KB
#define LDS_WHH_FLOATS (4 * HSZ * 16)
#define LDS_WIH_FLOATS (4 * ISZ * 16)
#define LDS_FLOATS     (LDS_WHH_FLOATS + LDS_WIH_FLOATS)

__device__ __forceinline__ float sigmoidf_(float x) {
    return 1.0f / (1.0f + __expf(-x));
}

// 8 consecutive k-steps (K += 32) of the 4-gate WMMA chain.
// a8[i] holds A[M=row][kbase+4i + 2*half .. +1]; bbase is the per-lane LDS base
// (lds_region + 32*half + l16); B element (g,k) sits at bbase + (g*KTOT + k)*16.
// b0..b3 are distinct so the dscnt waits can coalesce instead of serializing.
template <int KTOT>
__device__ __forceinline__ void wmma_x8(const v2f* __restrict__ a8,
                                        const float* __restrict__ bbase,
                                        int kbase, v8f* __restrict__ acc) {
    #pragma unroll
    for (int i = 0; i < 8; ++i) {
        const int k = kbase + 4 * i;
        const float* p0 = bbase + (0 * KTOT + k) * 16;
        const float* p1 = bbase + (1 * KTOT + k) * 16;
        const float* p2 = bbase + (2 * KTOT + k) * 16;
        const float* p3 = bbase + (3 * KTOT + k) * 16;
        v2f b0, b1, b2, b3;
        b0.x = p0[0]; b0.y = p0[16];
        b1.x = p1[0]; b1.y = p1[16];
        b2.x = p2[0]; b2.y = p2[16];
        b3.x = p3[0]; b3.y = p3[16];
        acc[0] = __builtin_amdgcn_wmma_f32_16x16x4_f32(false, a8[i], false, b0, (short)0, acc[0], false, false);
        acc[1] = __builtin_amdgcn_wmma_f32_16x16x4_f32(false, a8[i], false, b1, (short)0, acc[1], false, false);
        acc[2] = __builtin_amdgcn_wmma_f32_16x16x4_f32(false, a8[i], false, b2, (short)0, acc[2], false, false);
        acc[3] = __builtin_amdgcn_wmma_f32_16x16x4_f32(false, a8[i], false, b3, (short)0, acc[3], false, false);
    }
}

// Full K-dimension GEMM fragment with double-buffered batched A loads:
// 8 outstanding global_load_b64 per batch, overlapped with 32 WMMAs of the
// previous batch, so one L2 latency is amortized over 32 matrix ops.
template <int KTOT>
__device__ __forceinline__ void frag_gemm(const float* __restrict__ arow,
                                          const float* __restrict__ bbase,
                                          v8f* __restrict__ acc) {
    v2f a0[8], a1[8];
    #pragma unroll
    for (int i = 0; i < 8; ++i) a0[i] = *(const v2f*)(arow + 4 * i);
    #pragma unroll 1
    for (int kb = 0; kb < KTOT; kb += 64) {
        #pragma unroll
        for (int i = 0; i < 8; ++i) a1[i] = *(const v2f*)(arow + kb + 32 + 4 * i);
        wmma_x8<KTOT>(a0, bbase, kb, acc);
        if (kb + 64 < KTOT) {
            #pragma unroll
            for (int i = 0; i < 8; ++i) a0[i] = *(const v2f*)(arow + kb + 64 + 4 * i);
        }
        wmma_x8<KTOT>(a1, bbase, kb + 32, acc);
    }
}

// Zero the h ping-pong buffers (h0 = 0) and the barrier counter each call
// (harness poisons ws once and never re-poisons; we must be call-deterministic).
__global__ void lstm_init_kernel(float* __restrict__ hbuf, unsigned* __restrict__ cnt) {
    int i = blockIdx.x * blockDim.x + threadIdx.x;
    if (i < 2 * BATCH * HSZ) hbuf[i] = 0.0f;
    if (i == 0) *cnt = 0u;
}

__global__ __launch_bounds__(WG_THREADS, 1)
void lstm_persistent_kernel(const float* __restrict__ x,     // [B,S,I]
                            const float* __restrict__ wih,   // [I,4H]
                            const float* __restrict__ whh,   // [H,4H]
                            const float* __restrict__ bias,  // [4H]
                            float*       __restrict__ out,   // [B,S,H] ++ h_t[B,H] ++ c_t[B,H]
                            float*       __restrict__ hbuf,  // 2 * B*H ping-pong
                            unsigned*    __restrict__ cnt)   // grid barrier counter
{
    extern __shared__ float lds[];

    const int j0   = blockIdx.x * 16;       // hidden column block [j0, j0+16)
    const int tid  = threadIdx.x;
    const int lane = tid & 31;
    const int wave = tid >> 5;              // 0..3
    const int m0   = wave * 16;             // batch row tile
    const int half = lane >> 4;             // 0: K{0,1}/rows 0-7 ; 1: K{2,3}/rows 8-15
    const int l16  = lane & 15;

    // ---- stage this block's weight columns into LDS (read from HBM exactly once) ----
    for (int idx = tid; idx < LDS_FLOATS; idx += WG_THREADS) {
        float v;
        if (idx < LDS_WHH_FLOATS) {
            int g = idx >> 13;              // / (512*16)
            int k = (idx >> 4) & (HSZ - 1);
            int n = idx & 15;
            v = whh[(size_t)k * G4 + g * HSZ + j0 + n];
        } else {
            int id2 = idx - LDS_WHH_FLOATS;
            int g = id2 >> 12;              // / (256*16)
            int k = (id2 >> 4) & (ISZ - 1);
            int n = id2 & 15;
            v = wih[(size_t)k * G4 + g * HSZ + j0 + n];
        }
        lds[idx] = v;
    }
    __syncthreads();

    // per-lane LDS bases for B fragments: element (g,k) at base + (g*KTOT + k)*16
    const float* whh_b = lds + 32 * half + l16;
    const float* wih_b = lds + LDS_WHH_FLOATS + 32 * half + l16;

    // per-lane (one output column each) bias for the four gates
    float bias_g[4];
    #pragma unroll
    for (int g = 0; g < 4; ++g) bias_g[g] = bias[g * HSZ + j0 + l16];

    const v8f vzero = {};
    v8f c_frag = vzero;                     // cell-state tile: register resident

    const int    arow  = m0 + l16;                       // A-fragment row (batch index)
    const float* xrow0 = x + ((size_t)arow * SEQ) * ISZ + 2 * half;

    // prologue: input-GEMM for t = 0 (independent of h)
    v8f accx[4] = {};
    frag_gemm<ISZ>(xrow0, wih_b, accx);

    for (int t = 0; t < SEQ; ++t) {
        const float* hin  = hbuf + (t & 1) * (BATCH * HSZ);
        float*       hout = hbuf + ((t + 1) & 1) * (BATCH * HSZ);

        v8f acc[4];
        #pragma unroll
        for (int g = 0; g < 4; ++g) acc[g] = accx[g];

        // ---- recurrent contribution (the only thing on the sequential critical path) ----
        frag_gemm<HSZ>(hin + arow * HSZ + 2 * half, whh_b, acc);

        // ---- activations + state update (C/D layout: row = m0 + r + 8*half, col = j0+l16) ----
        v8f h_frag;
        #pragma unroll
        for (int r = 0; r < 8; ++r) {
            float ig = sigmoidf_(acc[0][r] + bias_g[0]);
            float fg = sigmoidf_(acc[1][r] + bias_g[1]);
            float gg = tanhf(acc[2][r] + bias_g[2]);
            float og = sigmoidf_(acc[3][r] + bias_g[3]);
            float cv = fg * c_frag[r] + ig * gg;
            c_frag[r] = cv;
            h_frag[r] = og * tanhf(cv);
        }

        // ---- publish h for the next step, then signal arrival ASAP ----
        #pragma unroll
        for (int r = 0; r < 8; ++r)
            hout[(m0 + r + 8 * half) * HSZ + j0 + l16] = h_frag[r];
        __threadfence();
        __syncthreads();
        if (tid == 0) atomicAdd(cnt, 1u);

        // ---- overlap window (hidden behind other WGs reaching the barrier) ----
        // hidden_sequence stores: nobody reads these, keep them off the critical path
        #pragma unroll
        for (int r = 0; r < 8; ++r) {
            int row = m0 + r + 8 * half;
            out[((size_t)row * SEQ + t) * HSZ + j0 + l16] = h_frag[r];
        }
        if (t == SEQ - 1) {
            size_t tail = (size_t)BATCH * SEQ * HSZ;
            #pragma unroll
            for (int r = 0; r < 8; ++r) {
                int row = m0 + r + 8 * half;
                out[tail + (size_t)row * HSZ + j0 + l16] = h_frag[r];                        // h_t
                out[tail + (size_t)BATCH * HSZ + (size_t)row * HSZ + j0 + l16] = c_frag[r];  // c_t
            }
        } else {
            // next step's input GEMM: no h dependency, so it hides inside the barrier wait
            if (t + 2 < SEQ)
                __builtin_prefetch(xrow0 + (size_t)(t + 2) * ISZ, 0, 1);
            #pragma unroll
            for (int g = 0; g < 4; ++g) accx[g] = vzero;
            frag_gemm<ISZ>(xrow0 + (size_t)(t + 1) * ISZ, wih_b, accx);
        }

        // ---- wait for all workgroups, then acquire their h stores ----
        if (tid == 0) {
            unsigned target = (unsigned)gridDim.x * (unsigned)(t + 1);
            while (atomicAdd(cnt, 0u) < target) {
                __builtin_amdgcn_s_sleep(2);
            }
        }
        __syncthreads();
        __threadfence();
    }
}

extern "C" void kernel_launch(void* const* d_in, const int* in_sizes, int n_in,
                              void* d_out, int out_size, void* d_ws, size_t ws_size,
                              hipStream_t stream) {
    const float* x    = (const float*)d_in[0];  // [B,S,I]
    const float* wih  = (const float*)d_in[1];  // [I,4H]
    const float* whh  = (const float*)d_in[2];  // [H,4H]
    const float* bias = (const float*)d_in[3];  // [4H]
    float*       out  = (float*)d_out;

    float*    hbuf = (float*)d_ws;                                        // 2*B*H floats
    unsigned* cnt  = (unsigned*)((char*)d_ws + 2 * BATCH * HSZ * sizeof(float));

    // reset h0=0 and barrier counter (deterministic per call)
    int initN = 2 * BATCH * HSZ;
    lstm_init_kernel<<<(initN + 255) / 256, 256, 0, stream>>>(hbuf, cnt);

    size_t shmem = LDS_FLOATS * sizeof(float);  // 192 KB (<= 320 KB CDNA5 WG limit)
    (void)hipFuncSetAttribute(reinterpret_cast<const void*>(lstm_persistent_kernel),
                              hipFuncAttributeMaxDynamicSharedMemorySize, (int)shmem);

    lstm_persistent_kernel<<<NWG, WG_THREADS, shmem, stream>>>(
        x, wih, whh, bias, out, hbuf, cnt);
}